// Quamb2Conv1D_40243843564243
// MI455X (gfx1250) — compile-verified
//
#include <hip/hip_runtime.h>
#include <hip/hip_bf16.h>
#include <stdint.h>

#define X_DIM    4096
#define X_HEADDIM  64
#define D_STATE   128
#define GD        128
#define D_TOTAL  (X_DIM + 2*GD)   /* 4352 */
#define BATCH       4
#define SEQLEN   4096

typedef uint32_t u32;
typedef uint64_t u64;
typedef __attribute__((ext_vector_type(4))) unsigned int v4u;
typedef __attribute__((ext_vector_type(4))) int          v4i;
typedef __attribute__((ext_vector_type(8))) int          v8i;
typedef __attribute__((ext_vector_type(4))) float        v4f;

#if __has_include(<hip/amd_detail/amd_gfx1250_TDM.h>)
#define TDM_ARITY 6
#else
#define TDM_ARITY 5
#endif

#if defined(__HIP_DEVICE_COMPILE__) && __has_builtin(__builtin_amdgcn_tensor_load_to_lds) && __has_builtin(__builtin_amdgcn_s_wait_tensorcnt)
#define USE_TDM 1
#else
#define USE_TDM 0
#endif

__device__ __forceinline__ int dot4_i8(u32 a, u32 b) {
#if defined(__HIP_DEVICE_COMPILE__) && __has_builtin(__builtin_amdgcn_sudot4)
  // V_DOT4_I32_IU8, both operands signed (NEG[0]=NEG[1]=1)
  return __builtin_amdgcn_sudot4(true, (int)a, true, (int)b, 0, false);
#elif defined(__HIP_DEVICE_COMPILE__) && __has_builtin(__builtin_amdgcn_sdot4)
  return __builtin_amdgcn_sdot4((int)a, (int)b, 0, false);
#else
  int s = 0;
#pragma unroll
  for (int i = 0; i < 4; ++i)
    s += (int)(int8_t)(a >> (8 * i)) * (int)(int8_t)(b >> (8 * i));
  return s;
#endif
}

__device__ __forceinline__ float fast_rcp(float x) {
#if defined(__HIP_DEVICE_COMPILE__) && __has_builtin(__builtin_amdgcn_rcpf)
  return __builtin_amdgcn_rcpf(x);      // v_rcp_f32, no Newton refinement
#else
  return 1.0f / x;
#endif
}

__device__ __forceinline__ float fast_exp2(float x) {
#if defined(__HIP_DEVICE_COMPILE__) && __has_builtin(__builtin_amdgcn_exp2f)
  return __builtin_amdgcn_exp2f(x);     // v_exp_f32
#else
  return exp2f(x);
#endif
}

__device__ __forceinline__ float clamp_q(float x) {
#if defined(__HIP_DEVICE_COMPILE__) && __has_builtin(__builtin_amdgcn_fmed3f)
  return __builtin_amdgcn_fmed3f(x, -128.0f, 127.0f);  // single v_med3_f32
#else
  return fminf(fmaxf(x, -128.0f), 127.0f);
#endif
}

__global__ __launch_bounds__(256) void quamb2_conv1d_kernel(
    const int8_t* __restrict__ xBC,
    const int8_t* __restrict__ weight,        // (D_TOTAL, 4) -> one dword per channel
    const int8_t* __restrict__ bias,
    const float*  __restrict__ x_out_scales,  // (1,4,4)
    const float*  __restrict__ B_out_scales,  // (1,)
    const float*  __restrict__ C_out_scales,  // (1,)
    const int*    __restrict__ head_range,    // (1,4)
    const int*    __restrict__ dim_range,     // (1,4,4)
    float*        __restrict__ out)
{
  // LDS layout: dword 0 = zero halo (causal left pad), row bytes at [4 .. 4+4096)
  __shared__ u32 lds[1 + SEQLEN / 4 + 3];

  const int tid = threadIdx.x;
  const int c   = blockIdx.x;   // channel
  const int b   = blockIdx.y;   // batch

  const size_t row_in = ((size_t)b * D_TOTAL + c) * (size_t)SEQLEN;

  if (tid == 0) lds[0] = 0u;   // causal halo

#if USE_TDM
  if (tid < 32) {   // one TDM issue per block (wave 0 only; branch skips other waves)
    const u64 gaddr  = (u64)(uintptr_t)(xBC + row_in);
    const u32 ldsRow = (u32)(uintptr_t)(&lds[1]);
    v4u g0;
    g0.x = 1u;                                              // count=1, user mode
    g0.y = ldsRow;                                          // lds_addr (bytes)
    g0.z = (u32)gaddr;                                      // global_addr[31:0]
    g0.w = ((u32)(gaddr >> 32) & 0x01FFFFFFu) | (2u << 30); // global_addr[56:32] | type=2
    v8i g1;
    g1[0] = 0;                         // wg_mask=0, data_size=1B, no flags
    g1[1] = (int)((u32)SEQLEN << 16);  // tensor_dim0[15:0] in [31:16]
    g1[2] = (int)(1u << 16);           // tensor_dim0 hi=0 ; tensor_dim1=1
    g1[3] = (int)((u32)SEQLEN << 16);  // tensor_dim1 hi=0 ; tile_dim0=4096
    g1[4] = 1;                         // tile_dim1=1, tile_dim2=0
    g1[5] = SEQLEN;                    // tensor_dim0_stride lo32
    g1[6] = 0;
    g1[7] = 0;
    v4i gz4 = {0, 0, 0, 0};
#if TDM_ARITY == 6
    v8i gz8 = {0, 0, 0, 0, 0, 0, 0, 0};
    __builtin_amdgcn_tensor_load_to_lds(g0, g1, gz4, gz4, gz8, 0);
#else
    __builtin_amdgcn_tensor_load_to_lds(g0, g1, gz4, gz4, 0);
#endif
    __builtin_amdgcn_s_wait_tensorcnt(0);
  }
  __syncthreads();
#else
  {
    const u32* gsrc = (const u32*)(xBC + row_in);
    u32 a0 = gsrc[tid * 4 + 0];
    u32 a1 = gsrc[tid * 4 + 1];
    u32 a2 = gsrc[tid * 4 + 2];
    u32 a3 = gsrc[tid * 4 + 3];
    lds[1 + tid * 4 + 0] = a0;
    lds[1 + tid * 4 + 1] = a1;
    lds[1 + tid * 4 + 2] = a2;
    lds[1 + tid * 4 + 3] = a3;
  }
  __syncthreads();
#endif

  // ---- per-channel (block-uniform -> SGPR) parameters ----
  float siw, sb, inv_out;
  if (c < X_DIM) {
    siw = 0.0123f * 0.008f;
    sb  = 0.004f;
    const int h = c / X_HEADDIM, dd = c % X_HEADDIM;
    int hg = (h >= head_range[0]) + (h >= head_range[1]) +
             (h >= head_range[2]) + (h >= head_range[3]);
    if (hg > 3) hg = 3;
    const int* dr = dim_range + hg * 4;
    int dg = (dd >= dr[0]) + (dd >= dr[1]) + (dd >= dr[2]) + (dd >= dr[3]);
    if (dg > 3) dg = 3;
    inv_out = fast_rcp(x_out_scales[hg * 4 + dg]);
  } else if (c < X_DIM + GD) {
    siw = 0.0117f * 0.009f;
    sb  = 0.0045f;
    inv_out = fast_rcp(B_out_scales[(c - X_DIM) / D_STATE]);
  } else {
    siw = 0.0131f * 0.0075f;
    sb  = 0.005f;
    inv_out = fast_rcp(C_out_scales[(c - X_DIM - GD) / D_STATE]);
  }
  const float bf  = (float)bias[c] * sb;
  const u32   wpk = ((const u32*)weight)[c];  // 4 taps packed, tap k in byte k

  // ---- output row base (x / B / C split) ----
  const size_t XTOT = (size_t)BATCH * X_DIM * SEQLEN;
  const size_t BTOT = (size_t)BATCH * GD * SEQLEN;
  size_t ob;
  if (c < X_DIM)            ob = ((size_t)b * X_DIM + c) * SEQLEN;
  else if (c < X_DIM + GD)  ob = XTOT + ((size_t)b * GD + (c - X_DIM)) * SEQLEN;
  else                      ob = XTOT + BTOT + ((size_t)b * GD + (c - X_DIM - GD)) * SEQLEN;
  float* orow = out + ob + (size_t)tid * 16;

  // ---- 16 outputs per thread: 5 LDS dwords -> funnel-shift windows -> dot4 ----
  u32 d[5];
#pragma unroll
  for (int j = 0; j < 5; ++j) d[j] = lds[tid * 4 + j];

  const float NLOG2E = -1.442695041f;   // -log2(e): silu(y) = y * rcp(1 + 2^(y*NLOG2E))

  float res[16];
#pragma unroll
  for (int g = 0; g < 4; ++g) {
    const u64 v = ((u64)d[g + 1] << 32) | (u64)d[g];
    u32 win[4];
    win[0] = (u32)(v >> 8);    // taps x[t-3..t] for t = 16*tid + 4g + 0
    win[1] = (u32)(v >> 16);
    win[2] = (u32)(v >> 24);
    win[3] = d[g + 1];
#pragma unroll
    for (int j = 0; j < 4; ++j) {
      const int   acc = dot4_i8(win[j], wpk);
      const float y   = __builtin_fmaf((float)acc, siw, bf);          // dequant conv + bias
      const float e   = fast_exp2(y * NLOG2E);                        // exp(-y)
      const float s   = y * fast_rcp(1.0f + e);                       // SiLU
      res[g * 4 + j]  = clamp_q(rintf(s * inv_out));                  // RNE quantize + clamp
    }
  }

  // write-once -> non-temporal 128-bit stores
#pragma unroll
  for (int g = 0; g < 4; ++g) {
    v4f o = { res[g * 4 + 0], res[g * 4 + 1], res[g * 4 + 2], res[g * 4 + 3] };
    __builtin_nontemporal_store(o, (v4f*)orow + g);
  }
}

extern "C" void kernel_launch(void* const* d_in, const int* in_sizes, int n_in,
                              void* d_out, int out_size, void* d_ws, size_t ws_size,
                              hipStream_t stream) {
  (void)in_sizes; (void)n_in; (void)out_size; (void)d_ws; (void)ws_size;
  dim3 grid(D_TOTAL, BATCH);   // one block per (channel, batch) row
  dim3 block(256);
  quamb2_conv1d_kernel<<<grid, block, 0, stream>>>(
      (const int8_t*)d_in[0],  // xBC
      (const int8_t*)d_in[1],  // weight
      (const int8_t*)d_in[2],  // bias
      (const float*)d_in[3],   // x_out_scales
      (const float*)d_in[4],   // B_out_scales
      (const float*)d_in[5],   // C_out_scales
      (const int*)d_in[6],     // x_head_group_range
      (const int*)d_in[7],     // x_dim_group_range
      (float*)d_out);
}